// DNN_Beamformer_34677565948213
// MI455X (gfx1250) — compile-verified
//
#include <hip/hip_runtime.h>
#include <math.h>

typedef __attribute__((ext_vector_type(2))) float v2f;
typedef __attribute__((ext_vector_type(8))) float v8f;

constexpr int kB = 8;
constexpr int kT = 512;
constexpr int kC = 8;
constexpr int kF = 257;
constexpr int kA = 320;

constexpr int kFT = 16;                       // f-tile width per block
constexpr int kCT = 64;                       // t-chunk per LDS stage
constexpr int kNFT = (kF + kFT - 1) / kFT;    // 17 f-tiles
constexpr int kXPitch = 17;                   // padded f pitch (gcd(17,64)=1)

__device__ __forceinline__ float2 cmul(float2 a, float2 b) {
    return make_float2(a.x * b.x - a.y * b.y, a.x * b.y + a.y * b.x);
}

// ---------------------------------------------------------------------------
// Kernel 1: masked PSD (speech + noise) via f32 WMMA 16x16x4.
// Block = (b, 16-wide f-tile). Data + masks staged through LDS with fully
// coalesced loads -> each input element read from HBM exactly once.
// A rows 0-7 = m*Re(x_c), rows 8-15 = m*Im(x_c); B cols mirror unmasked.
// 16x16 result blocks: Re = RR+II, Im = IR-RI (conj folded into combination).
// Each wave owns 2 f's for the whole T range -> accumulators stay in VGPRs.
// ---------------------------------------------------------------------------
__global__ void psd_kernel(const float* __restrict__ dr, const float* __restrict__ di,
                           const float* __restrict__ ms, const float* __restrict__ mn,
                           float2* __restrict__ psd_s, float2* __restrict__ psd_n) {
    __shared__ float mS[kFT * kT];                     // 32 KB
    __shared__ float mN[kFT * kT];                     // 32 KB
    __shared__ float xs[kCT * 2 * kC * kXPitch];       // 68 KB (also reused as dump)
    __shared__ float scaleS[kFT], scaleN[kFT];

    const int f0 = blockIdx.x * kFT;
    const int b = blockIdx.y;
    const int tid = threadIdx.x;

    // ---- phase 1: channel-mean masks (coalesced over t), normalize over T --
    for (int idx = tid; idx < kFT * kT; idx += 256) {
        const int fl = idx >> 9;              // /512
        const int t = idx & (kT - 1);
        const int f = min(f0 + fl, kF - 1);
        const size_t base = ((size_t)(b * kF + f) * kC) * kT + t;
        float ss = 0.f, sn = 0.f;
        for (int c = 0; c < kC; ++c) { ss += ms[base + c * kT]; sn += mn[base + c * kT]; }
        mS[idx] = ss * (1.f / kC);
        mN[idx] = sn * (1.f / kC);
    }
    __syncthreads();
    if (tid < 32) {
        const int fl = tid & 15;
        const float* src = (tid < 16) ? &mS[fl * kT] : &mN[fl * kT];
        float s = 0.f;
        for (int i = 0; i < kT; ++i) s += src[(i + fl * 33) & (kT - 1)];  // bank-skewed
        const float inv = 1.f / (s + 1e-15f);
        if (tid < 16) scaleS[fl] = inv; else scaleN[fl] = inv;
    }
    __syncthreads();
    for (int idx = tid; idx < kFT * kT; idx += 256) {
        mS[idx] *= scaleS[idx >> 9];
        mN[idx] *= scaleN[idx >> 9];
    }

    // ---- phase 2: t-chunked staged WMMA accumulation -----------------------
    const int wave = tid >> 5, lane = tid & 31;
    const int M = lane & 15;
    const int c = M & 7, part = M >> 3;       // rows 0-7 real, 8-15 imag
    const int tOff = (lane >= 16) ? 2 : 0;    // upper lanes hold K=2,3

    v8f aS0 = {}, aN0 = {}, aS1 = {}, aN1 = {};

    for (int ch = 0; ch < kT / kCT; ++ch) {
        __syncthreads();  // previous chunk's compute done before overwrite
        // stage data chunk: (t_local, part, c, fl); 16 lanes read 64B rows
        for (int idx = tid; idx < kCT * 2 * kC * kFT; idx += 256) {
            const int fl = idx & 15;
            const int cc = (idx >> 4) & 7;
            const int pp = (idx >> 7) & 1;
            const int tl = idx >> 8;
            const int f = min(f0 + fl, kF - 1);
            const int t = ch * kCT + tl;
            const float* src = pp ? di : dr;
            xs[((tl * 2 + pp) * kC + cc) * kXPitch + fl] =
                src[((size_t)(b * kT + t) * kC + cc) * kF + f];
        }
        __syncthreads();

        const int tg = ch * kCT;
        for (int k = 0; k < kCT; k += 4) {
            const int ta = k + tOff;
            const int tb = ta + 1;
            const int rowA = ((ta * 2 + part) * kC + c) * kXPitch;
            const int rowB = ((tb * 2 + part) * kC + c) * kXPitch;
#pragma unroll
            for (int fs = 0; fs < 2; ++fs) {
                const int fl = wave * 2 + fs;
                const float xa = xs[rowA + fl];
                const float xb = xs[rowB + fl];
                v2f Bv; Bv[0] = xa;                      Bv[1] = xb;
                v2f As; As[0] = mS[fl * kT + tg + ta] * xa; As[1] = mS[fl * kT + tg + tb] * xb;
                v2f An; An[0] = mN[fl * kT + tg + ta] * xa; An[1] = mN[fl * kT + tg + tb] * xb;
                if (fs == 0) {
                    aS0 = __builtin_amdgcn_wmma_f32_16x16x4_f32(false, As, false, Bv, (short)0, aS0, false, false);
                    aN0 = __builtin_amdgcn_wmma_f32_16x16x4_f32(false, An, false, Bv, (short)0, aN0, false, false);
                } else {
                    aS1 = __builtin_amdgcn_wmma_f32_16x16x4_f32(false, As, false, Bv, (short)0, aS1, false, false);
                    aN1 = __builtin_amdgcn_wmma_f32_16x16x4_f32(false, An, false, Bv, (short)0, aN1, false, false);
                }
            }
        }
    }

    // ---- phase 3: dump accumulators into (reused) xs, combine, store -------
    __syncthreads();
    float* dump = xs;  // need 8 waves * 4 accs * 256 floats = 8192 <= 17408
#pragma unroll
    for (int j = 0; j < 8; ++j) {
        dump[((wave * 4 + 0) * 8 + j) * 32 + lane] = aS0[j];
        dump[((wave * 4 + 1) * 8 + j) * 32 + lane] = aN0[j];
        dump[((wave * 4 + 2) * 8 + j) * 32 + lane] = aS1[j];
        dump[((wave * 4 + 3) * 8 + j) * 32 + lane] = aN1[j];
    }
    __syncthreads();

    // 16 fl * 2 matrices * 64 complex entries = 2048 outputs
    for (int idx = tid; idx < kFT * 2 * 64; idx += 256) {
        const int fl = idx >> 7;
        const int rem = idx & 127;
        const int mat = rem >> 6;       // 0 = speech, 1 = noise
        const int ce = rem & 63;
        const int cc = ce >> 3, ee = ce & 7;
        const int wv = fl >> 1, fs = fl & 1;
        const float* dbl = &dump[((wv * 4 + fs * 2 + mat) * 8) * 32];
        // element (M,N): vgpr j = M&7, lane = (M>=8 ? 16 : 0) + N
        const float RR = dbl[cc * 32 + ee];           // (c,   e)
        const float II = dbl[cc * 32 + 24 + ee];      // (8+c, 8+e)
        const float IR = dbl[cc * 32 + 16 + ee];      // (8+c, e)
        const float RI = dbl[cc * 32 + 8 + ee];       // (c,   8+e)
        const int f = f0 + fl;
        if (f < kF) {
            float2* dst = (mat == 0) ? psd_s : psd_n;
            dst[(size_t)(b * kF + f) * 64 + ce] = make_float2(RR + II, IR - RI);
        }
    }
}

// ---------------------------------------------------------------------------
// Kernel 2a: psd feature |sum_{e!=c} psd_s[b,f,c,e]| / (C-1) -> (B,C,F)
// ---------------------------------------------------------------------------
__global__ void feat_kernel(const float2* __restrict__ psd_s, float* __restrict__ feat) {
    const int idx = blockIdx.x * 256 + threadIdx.x;
    if (idx >= kB * kC * kF) return;
    const int b = idx / (kC * kF);
    const int rem = idx % (kC * kF);
    const int c = rem / kF, f = rem % kF;
    const float2* p = psd_s + (size_t)(b * kF + f) * 64 + c * 8;
    float sr = 0.f, si = 0.f;
    for (int e = 0; e < kC; ++e) {
        if (e == c) continue;
        float2 v = p[e];
        sr += v.x;
        si += v.y;
    }
    sr *= (1.f / (kC - 1));
    si *= (1.f / (kC - 1));
    feat[(size_t)(b * kC + c) * kF + f] = sqrtf(sr * sr + si * si);
}

// ---------------------------------------------------------------------------
// Kernel 2b: MLP GEMM (64 x 257) @ (257 x 320) + bias, tanh, via f32 WMMA.
// One wave per 16x16 output tile (4 x 20 tiles).
// ---------------------------------------------------------------------------
__global__ void mlp_gemm_kernel(const float* __restrict__ feat, const float* __restrict__ W,
                                const float* __restrict__ bias, float* __restrict__ mlp) {
    const int tile = blockIdx.x;
    const int mt = tile / 20, nt = tile % 20;
    const int lane = threadIdx.x;
    const int r = mt * 16 + (lane & 15);
    const int n = nt * 16 + (lane & 15);
    const int tOff = (lane >= 16) ? 2 : 0;
    v8f acc = {};
    for (int k0 = 0; k0 < kF; k0 += 4) {
        const int ka = k0 + tOff, kb = ka + 1;
        v2f Av, Bv;
        Av[0] = (ka < kF) ? feat[(size_t)r * kF + ka] : 0.f;
        Av[1] = (kb < kF) ? feat[(size_t)r * kF + kb] : 0.f;
        Bv[0] = (ka < kF) ? W[(size_t)ka * kA + n] : 0.f;
        Bv[1] = (kb < kF) ? W[(size_t)kb * kA + n] : 0.f;
        acc = __builtin_amdgcn_wmma_f32_16x16x4_f32(false, Av, false, Bv, (short)0, acc, false, false);
    }
    const int rowBase = mt * 16 + ((lane >= 16) ? 8 : 0);
    const int col = nt * 16 + (lane & 15);
    for (int j = 0; j < 8; ++j) {
        mlp[(size_t)(rowBase + j) * kA + col] = tanhf(acc[j] + bias[col]);
    }
}

// ---------------------------------------------------------------------------
// Kernel 2c: gvec projection + softmax(2*e) over channels -> u (B,C)
// ---------------------------------------------------------------------------
__global__ void attn_kernel(const float* __restrict__ mlp, const float* __restrict__ gw,
                            const float* __restrict__ gb, float* __restrict__ u) {
    __shared__ float sh[64];
    const int tid = threadIdx.x;  // 64 threads = (b,c) pairs
    float e = gb[0];
    for (int a = 0; a < kA; ++a) e += mlp[(size_t)tid * kA + a] * gw[a];
    sh[tid] = 2.0f * e;
    __syncthreads();
    const int b = tid >> 3;
    float mx = -1e30f;
    for (int j = 0; j < 8; ++j) mx = fmaxf(mx, sh[b * 8 + j]);
    float sum = 0.f;
    for (int j = 0; j < 8; ++j) sum += __expf(sh[b * 8 + j] - mx);
    u[tid] = __expf(sh[tid] - mx) / sum;
}

// ---------------------------------------------------------------------------
// Kernel 3: MVDR weights. Per (b,f): inv(psd_n) via Gauss-Jordan w/ partial
// pivoting (8x8 complex, per-lane matrices in padded LDS rows), then
// ws[e] = (sum_k inv[e,k] * (sum_c S[k,c] u[c])) / (trace + eps).
// ---------------------------------------------------------------------------
__global__ void mvdr_kernel(const float2* __restrict__ psd_s, const float2* __restrict__ psd_n,
                            const float* __restrict__ u, float2* __restrict__ wsv) {
    __shared__ float2 Am[32][65];  // padded row: 65*8B = 520B -> conflict-free
    __shared__ float2 Iv[32][65];
    const int tid = threadIdx.x;
    const int idx = blockIdx.x * 32 + tid;
    if (idx >= kB * kF) return;
    const int b = idx / kF;
    const float2* Np = psd_n + (size_t)idx * 64;
    const float2* Sp = psd_s + (size_t)idx * 64;
    float2(&Ar)[65] = Am[tid];
    float2(&Ir)[65] = Iv[tid];
    for (int i = 0; i < 64; ++i) {
        Ar[i] = Np[i];
        Ir[i] = make_float2((i % 9 == 0) ? 1.f : 0.f, 0.f);
    }
    for (int k = 0; k < 8; ++k) {
        int p = k;
        float2 d0 = Ar[k * 8 + k];
        float best = d0.x * d0.x + d0.y * d0.y;
        for (int r = k + 1; r < 8; ++r) {
            float2 v = Ar[r * 8 + k];
            float m2 = v.x * v.x + v.y * v.y;
            if (m2 > best) { best = m2; p = r; }
        }
        if (p != k) {
            for (int j = 0; j < 8; ++j) {
                float2 t1 = Ar[k * 8 + j]; Ar[k * 8 + j] = Ar[p * 8 + j]; Ar[p * 8 + j] = t1;
                float2 t2 = Ir[k * 8 + j]; Ir[k * 8 + j] = Ir[p * 8 + j]; Ir[p * 8 + j] = t2;
            }
        }
        float2 piv = Ar[k * 8 + k];
        float inv_d = 1.f / (piv.x * piv.x + piv.y * piv.y);
        float2 pinv = make_float2(piv.x * inv_d, -piv.y * inv_d);
        for (int j = 0; j < 8; ++j) {
            Ar[k * 8 + j] = cmul(Ar[k * 8 + j], pinv);
            Ir[k * 8 + j] = cmul(Ir[k * 8 + j], pinv);
        }
        for (int r = 0; r < 8; ++r) {
            if (r == k) continue;
            float2 fct = Ar[r * 8 + k];
            for (int j = 0; j < 8; ++j) {
                float2 a = cmul(fct, Ar[k * 8 + j]);
                Ar[r * 8 + j].x -= a.x;
                Ar[r * 8 + j].y -= a.y;
                float2 ii = cmul(fct, Ir[k * 8 + j]);
                Ir[r * 8 + j].x -= ii.x;
                Ir[r * 8 + j].y -= ii.y;
            }
        }
    }
    float uv[8];
    for (int cc = 0; cc < 8; ++cc) uv[cc] = u[b * 8 + cc];
    float2 su[8];
    for (int k = 0; k < 8; ++k) {
        float2 s = make_float2(0.f, 0.f);
        for (int cc = 0; cc < 8; ++cc) {
            float2 v = Sp[k * 8 + cc];
            s.x += v.x * uv[cc];
            s.y += v.y * uv[cc];
        }
        su[k] = s;
    }
    float2 tr = make_float2(1e-15f, 0.f);  // trace + EPS
    for (int i = 0; i < 8; ++i)
        for (int k = 0; k < 8; ++k) {
            float2 p2 = cmul(Ir[i * 8 + k], Sp[k * 8 + i]);
            tr.x += p2.x;
            tr.y += p2.y;
        }
    const float td = 1.f / (tr.x * tr.x + tr.y * tr.y);
    const float2 trinv = make_float2(tr.x * td, -tr.y * td);
    for (int e = 0; e < 8; ++e) {
        float2 v = make_float2(0.f, 0.f);
        for (int k = 0; k < 8; ++k) {
            float2 p2 = cmul(Ir[e * 8 + k], su[k]);
            v.x += p2.x;
            v.y += p2.y;
        }
        wsv[(size_t)idx * 8 + e] = cmul(v, trinv);
    }
}

// ---------------------------------------------------------------------------
// Kernel 4: apply beamformer: out[b,t,f] = sum_c conj(ws[b,f,c]) * x[b,f,c,t]
// 288 threads cover all F=257 in one block; 4 t per block amortize ws loads.
// f contiguous across lanes -> fully coalesced; vectorized float2 store.
// ---------------------------------------------------------------------------
__global__ void beamform_kernel(const float* __restrict__ dr, const float* __restrict__ di,
                                const float2* __restrict__ wsv, float* __restrict__ out) {
    const int f = threadIdx.x;
    if (f >= kF) return;
    const int b = blockIdx.y;
    const int t0 = blockIdx.x * 4;
    float2 w[8];
#pragma unroll
    for (int c = 0; c < 8; ++c) w[c] = wsv[(size_t)(b * kF + f) * 8 + c];
    float2* out2 = (float2*)out;
    for (int dt = 0; dt < 4; ++dt) {
        const int t = t0 + dt;
        const size_t dbase = ((size_t)(b * kT + t) * kC) * kF + f;
        float ar = 0.f, ai = 0.f;
#pragma unroll
        for (int c = 0; c < 8; ++c) {
            const float xr = dr[dbase + (size_t)c * kF];
            const float xi = di[dbase + (size_t)c * kF];
            ar += w[c].x * xr + w[c].y * xi;   // conj(w) * x
            ai += w[c].x * xi - w[c].y * xr;
        }
        out2[(size_t)(b * kT + t) * kF + f] = make_float2(ar, ai);
    }
}

extern "C" void kernel_launch(void* const* d_in, const int* in_sizes, int n_in,
                              void* d_out, int out_size, void* d_ws, size_t ws_size,
                              hipStream_t stream) {
    (void)in_sizes; (void)n_in; (void)out_size; (void)ws_size;
    const float* dr     = (const float*)d_in[0];
    const float* di     = (const float*)d_in[1];
    const float* ms     = (const float*)d_in[2];
    const float* mn     = (const float*)d_in[3];
    const float* mlp_w  = (const float*)d_in[4];
    const float* mlp_b  = (const float*)d_in[5];
    const float* gvec_w = (const float*)d_in[6];
    const float* gvec_b = (const float*)d_in[7];
    float* out = (float*)d_out;

    char* ws = (char*)d_ws;
    size_t off = 0;
    float2* psd_s = (float2*)(ws + off); off += (size_t)kB * kF * 64 * sizeof(float2);
    float2* psd_n = (float2*)(ws + off); off += (size_t)kB * kF * 64 * sizeof(float2);
    float*  feat  = (float*)(ws + off);  off += (size_t)kB * kC * kF * sizeof(float);
    float*  mlp   = (float*)(ws + off);  off += (size_t)kB * kC * kA * sizeof(float);
    float*  u     = (float*)(ws + off);  off += (size_t)kB * kC * sizeof(float);
    float2* wsv   = (float2*)(ws + off); off += (size_t)kB * kF * kC * sizeof(float2);

    psd_kernel<<<dim3(kNFT, kB), 256, 0, stream>>>(dr, di, ms, mn, psd_s, psd_n);
    feat_kernel<<<(kB * kC * kF + 255) / 256, 256, 0, stream>>>(psd_s, feat);
    mlp_gemm_kernel<<<(64 / 16) * (kA / 16), 32, 0, stream>>>(feat, mlp_w, mlp_b, mlp);
    attn_kernel<<<1, 64, 0, stream>>>(mlp, gvec_w, gvec_b, u);
    mvdr_kernel<<<(kB * kF + 31) / 32, 32, 0, stream>>>(psd_s, psd_n, u, wsv);
    beamform_kernel<<<dim3(kT / 4, kB), 288, 0, stream>>>(dr, di, wsv, out);
}